// PGALoss_55130200212218
// MI455X (gfx1250) — compile-verified
//
#include <hip/hip_runtime.h>

typedef __attribute__((ext_vector_type(2))) float v2f;
typedef __attribute__((ext_vector_type(8))) float v8f;

#define WAVES_PER_BLOCK 8
#define TILE 16

// acc[0] = sum over source points of (sqrt(1+d2)+d), d = dist to nearest target
// acc[1] = same with roles swapped
__global__ void pga_init_acc(float* acc) {
    if (threadIdx.x < 2) acc[threadIdx.x] = 0.0f;
}

// Pack (B,N,3) raw points into (B,N,4) = (x, y, z, -|p|^2/2): exactly the
// per-lane WMMA B-operand layout (half 0 -> K0,K1 ; half 1 -> K2,K3).
__global__ __launch_bounds__(256) void pga_pack_kernel(
    const float* __restrict__ src, const float* __restrict__ tgt,
    float* __restrict__ padS, float* __restrict__ padT, int totalPts)
{
    const int t = blockIdx.x * blockDim.x + threadIdx.x;   // 0 .. 2*B*N-1
    if (t >= 2 * totalPts) return;
    const int arr = t >= totalPts;                         // 0=src, 1=tgt
    const int j = arr ? (t - totalPts) : t;
    const float* in = (arr ? tgt : src) + (size_t)j * 3;
    float* out = (arr ? padT : padS) + (size_t)j * 4;
    const float x = in[0], y = in[1], z = in[2];
    out[0] = x;
    out[1] = y;
    out[2] = z;
    out[3] = -0.5f * (x * x + y * y + z * z);
}

// Fast path: operands pre-packed, inner loop = b64 load + WMMA + max3.
__global__ __launch_bounds__(256) void pga_nn_loss_packed(
    const float* __restrict__ padS, const float* __restrict__ padT,
    float* __restrict__ acc, int N, int nQt)
{
    const int lane = threadIdx.x & 31;
    const int wave = blockIdx.x * WAVES_PER_BLOCK + (threadIdx.x >> 5);

    const int perDir = 2 * nQt;             // B = 2
    const int dir   = wave / perDir;
    const int rem   = wave - dir * perDir;
    const int b     = rem / nQt;
    const int qtile = rem - b * nQt;

    const float* __restrict__ Q = (dir == 0 ? padS : padT) + (size_t)b * N * 4;
    const float* __restrict__ P = (dir == 0 ? padT : padS) + (size_t)b * N * 4;

    const int half = lane >> 4;   // 0: K=0,1 lanes  1: K=2,3 lanes
    const int li   = lane & 15;
    const int koff = 2 * half;

    // A matrix: 16 queries x K=4, K = (x, y, z, 1). The K=3 "1" multiplies
    // B's K=3 slot (-|p|^2/2):  D[m][n] = q_m . p_n - |p_n|^2/2
    const int qi = qtile * TILE + li;
    const v2f aq = *(const v2f*)(Q + 4 * qi + koff);
    v2f a;
    a.x = aq.x;                   // x (half 0) / z (half 1)
    a.y = half ? 1.0f : aq.y;     // y (half 0) / 1.0 (half 1)

    // Query norms for this lane's 8 C/D rows (row M = half*8 + r)
    float qn[8];
#pragma unroll
    for (int r = 0; r < 8; ++r) {
        const int m = qtile * TILE + half * 8 + r;
        qn[r] = -2.0f * Q[4 * m + 3];
    }

    // Running max of D;  min dist^2 = qn - 2*max(D)
    float vmax[8];
#pragma unroll
    for (int r = 0; r < 8; ++r) vmax[r] = -3.4e38f;

    const float* Pl = P + koff;            // this lane's K-pair base
    const int nCt = N / TILE;
    for (int ct = 0; ct < nCt; ++ct) {
        const v2f bv = *(const v2f*)(Pl + 4 * (ct * TILE + li));  // b64 load

        v8f c = {0.f, 0.f, 0.f, 0.f, 0.f, 0.f, 0.f, 0.f};
        c = __builtin_amdgcn_wmma_f32_16x16x4_f32(
                false, a, false, bv, (short)0, c, false, false);

#pragma unroll
        for (int r = 0; r < 8; ++r)
            vmax[r] = fmaxf(vmax[r], c[r]);   // fuses into v_max3_num_f32
    }

#pragma unroll
    for (int r = 0; r < 8; ++r) {
        float v = vmax[r];
        v = fmaxf(v, __shfl_xor(v, 1, 32));
        v = fmaxf(v, __shfl_xor(v, 2, 32));
        v = fmaxf(v, __shfl_xor(v, 4, 32));
        v = fmaxf(v, __shfl_xor(v, 8, 32));
        vmax[r] = v;
    }

    float sum = 0.0f;
    if (li == 0) {
#pragma unroll
        for (int r = 0; r < 8; ++r) {
            const float d2 = fmaxf(fmaf(-2.0f, vmax[r], qn[r]), 0.0f);
            sum += sqrtf(1.0f + d2) + sqrtf(d2);   // ||gp|| + ||dual||
        }
    }
    sum += __shfl_down(sum, 16, 32);
    if (lane == 0) atomicAdd(&acc[dir], sum);
}

// Fallback (small ws_size): build operands from raw points in-loop.
__global__ __launch_bounds__(256) void pga_nn_loss_raw(
    const float* __restrict__ src, const float* __restrict__ tgt,
    float* __restrict__ acc, int N, int nQt)
{
    const int lane = threadIdx.x & 31;
    const int wave = blockIdx.x * WAVES_PER_BLOCK + (threadIdx.x >> 5);
    const int perDir = 2 * nQt;
    const int dir   = wave / perDir;
    const int rem   = wave - dir * perDir;
    const int b     = rem / nQt;
    const int qtile = rem - b * nQt;

    const float* __restrict__ q = (dir == 0 ? src : tgt) + (size_t)b * N * 3;
    const float* __restrict__ p = (dir == 0 ? tgt : src) + (size_t)b * N * 3;
    const int half = lane >> 4;
    const int li   = lane & 15;

    const int qi = qtile * TILE + li;
    const float qx = q[3 * qi], qy = q[3 * qi + 1], qz = q[3 * qi + 2];
    v2f a;
    a.x = half ? qz : qx;
    a.y = half ? 1.0f : qy;

    float qn[8];
#pragma unroll
    for (int r = 0; r < 8; ++r) {
        const int m = qtile * TILE + half * 8 + r;
        const float x = q[3 * m], y = q[3 * m + 1], z = q[3 * m + 2];
        qn[r] = x * x + y * y + z * z;
    }

    float vmax[8];
#pragma unroll
    for (int r = 0; r < 8; ++r) vmax[r] = -3.4e38f;

    const int nCt = N / TILE;
    for (int ct = 0; ct < nCt; ++ct) {
        const int cj = ct * TILE + li;
        const float px = p[3 * cj], py = p[3 * cj + 1], pz = p[3 * cj + 2];
        const float pnh = -0.5f * (px * px + py * py + pz * pz);
        v2f bv;
        bv.x = half ? pz : px;
        bv.y = half ? pnh : py;
        v8f c = {0.f, 0.f, 0.f, 0.f, 0.f, 0.f, 0.f, 0.f};
        c = __builtin_amdgcn_wmma_f32_16x16x4_f32(
                false, a, false, bv, (short)0, c, false, false);
#pragma unroll
        for (int r = 0; r < 8; ++r) vmax[r] = fmaxf(vmax[r], c[r]);
    }

#pragma unroll
    for (int r = 0; r < 8; ++r) {
        float v = vmax[r];
        v = fmaxf(v, __shfl_xor(v, 1, 32));
        v = fmaxf(v, __shfl_xor(v, 2, 32));
        v = fmaxf(v, __shfl_xor(v, 4, 32));
        v = fmaxf(v, __shfl_xor(v, 8, 32));
        vmax[r] = v;
    }

    float sum = 0.0f;
    if (li == 0) {
#pragma unroll
        for (int r = 0; r < 8; ++r) {
            const float d2 = fmaxf(fmaf(-2.0f, vmax[r], qn[r]), 0.0f);
            sum += sqrtf(1.0f + d2) + sqrtf(d2);
        }
    }
    sum += __shfl_down(sum, 16, 32);
    if (lane == 0) atomicAdd(&acc[dir], sum);
}

__global__ void pga_finalize(const float* __restrict__ acc,
                             float* __restrict__ out, float invBN)
{
    const float l0 = acc[0] * invBN;
    const float l1 = acc[1] * invBN;
    const float c0 = fminf(fmaxf((l0 - 1.0f) * 0.5f, 0.0f), 1.0f);
    const float c1 = fminf(fmaxf((l1 - 1.0f) * 0.5f, 0.0f), 1.0f);
    out[0] = 0.5f * (c0 + c1);
}

extern "C" void kernel_launch(void* const* d_in, const int* in_sizes, int n_in,
                              void* d_out, int out_size, void* d_ws, size_t ws_size,
                              hipStream_t stream) {
    (void)n_in; (void)out_size;
    const float* src = (const float*)d_in[0];   // (B,N,3) f32
    const float* tgt = (const float*)d_in[1];   // (B,N,3) f32
    const int B = 2;
    const int N = in_sizes[0] / (B * 3);        // 8192
    const int nQt = N / TILE;                   // 512
    const int totalPts = B * N;

    float* acc = (float*)d_ws;                  // 2 floats
    pga_init_acc<<<1, 64, 0, stream>>>(acc);

    const int totalWaves = 2 * B * nQt;         // 2048 -> exact grid, EXEC all-1s
    const int blocks = totalWaves / WAVES_PER_BLOCK;

    const size_t padBytes = (size_t)totalPts * 4 * sizeof(float);   // per array
    const size_t need = 256 + 2 * padBytes;

    if (ws_size >= need) {
        float* padS = (float*)((char*)d_ws + 256);
        float* padT = padS + (size_t)totalPts * 4;
        const int packThreads = 2 * totalPts;
        pga_pack_kernel<<<(packThreads + 255) / 256, 256, 0, stream>>>(
            src, tgt, padS, padT, totalPts);
        pga_nn_loss_packed<<<blocks, WAVES_PER_BLOCK * 32, 0, stream>>>(
            padS, padT, acc, N, nQt);
    } else {
        pga_nn_loss_raw<<<blocks, WAVES_PER_BLOCK * 32, 0, stream>>>(
            src, tgt, acc, N, nQt);
    }

    pga_finalize<<<1, 1, 0, stream>>>(acc, (float*)d_out, 1.0f / (float)(B * N));
}